// DigitOutputLayers_51754355917418
// MI455X (gfx1250) — compile-verified
//
#include <hip/hip_runtime.h>
#include <stdint.h>

#define K_CLS   10
#define R_PROP  500000
#define NCAND   (R_PROP * K_CLS)
#define T_TOP   2048
#define CAP     8192
#define HBINS   262144
#define SCORE_T 0.05f
#define NMS_T   0.5f
#define IMG_W   1920.0f
#define IMG_H   1080.0f
#define CLS_OFF 3001.0f   /* IMG_W + IMG_H + 1 */

typedef float v2f  __attribute__((ext_vector_type(2)));
typedef float v8f  __attribute__((ext_vector_type(8)));
typedef unsigned int u32x4 __attribute__((ext_vector_type(4)));
typedef int i32x4 __attribute__((ext_vector_type(4)));
typedef int i32x8 __attribute__((ext_vector_type(8)));

/* ---- workspace layout (bytes) ---- */
#define WS_HIST 0u
#define WS_CNT  (WS_HIST + HBINS * 4u)            /* u32[16]: [0]=gather count, [1]=thr bits */
#define WS_CAND (WS_CNT + 64u)                    /* u64[CAP] */
#define WS_SSC  (WS_CAND + CAP * 8u)              /* f32[T]   selected scores */
#define WS_SBOX (WS_SSC + T_TOP * 4u)             /* f32[T*4] clipped boxes   */
#define WS_NBOX (WS_SBOX + T_TOP * 16u)           /* f32[T*4] +class offset   */
#define WS_SUP  (WS_NBOX + T_TOP * 16u)           /* u32[T*128] suppression   */

/* ------------------------------------------------------------------ */
__global__ void k_zero(unsigned* p, int n) {
  int i = blockIdx.x * blockDim.x + threadIdx.x;
  if (i < n) p[i] = 0u;
}

/* ------------------------------------------------------------------ */
__global__ void k_hist(const float* __restrict__ scores,
                       const float* __restrict__ center,
                       unsigned* __restrict__ hist) {
  int idx = blockIdx.x * blockDim.x + threadIdx.x;
  if (idx >= NCAND) return;
  int r = idx / K_CLS, c = idx % K_CLS;
  float s = scores[r * (K_CLS + 1) + 1 + c] * center[r];
  if (s > SCORE_T) atomicAdd(&hist[__float_as_uint(s) >> 12], 1u);
}

/* ------------------------------------------------------------------ */
__global__ __launch_bounds__(1024) void k_thresh(const unsigned* __restrict__ hist,
                                                 unsigned* __restrict__ cnts) {
  __shared__ unsigned chunk[1024];
  __shared__ unsigned red[1024];
  __shared__ unsigned s_cum, s_found, s_thr;
  int tid = threadIdx.x;
  if (tid == 0) { s_cum = 0u; s_found = 0u; s_thr = 0u; }
  __syncthreads();
  for (int base = HBINS - 1024; base >= 0; base -= 1024) {
    if (s_found) break;
    unsigned v = hist[base + tid];
    chunk[tid] = v; red[tid] = v;
    __syncthreads();
    for (int off = 512; off > 0; off >>= 1) {
      if (tid < off) red[tid] += red[tid + off];
      __syncthreads();
    }
    unsigned csum = red[0];
    if (tid == 0) {
      if (s_cum + csum >= (unsigned)T_TOP) {
        unsigned cum = s_cum;
        for (int k = 1023; k >= 0; --k) {
          unsigned c = chunk[k];
          if (cum + c >= (unsigned)T_TOP) { s_thr = ((unsigned)(base + k)) << 12; s_found = 1u; break; }
          cum += c;
        }
      } else {
        s_cum += csum;
      }
    }
    __syncthreads();
  }
  if (tid == 0) cnts[1] = s_thr; /* 0 => fewer than T valid: take everything > SCORE_T */
}

/* ------------------------------------------------------------------ */
__global__ void k_gather(const float* __restrict__ scores,
                         const float* __restrict__ center,
                         unsigned* __restrict__ cnts,
                         unsigned long long* __restrict__ cand) {
  int idx = blockIdx.x * blockDim.x + threadIdx.x;
  if (idx >= NCAND) return;
  int r = idx / K_CLS, c = idx % K_CLS;
  float s = scores[r * (K_CLS + 1) + 1 + c] * center[r];
  if (s > SCORE_T) {
    unsigned bits = __float_as_uint(s);
    if (bits >= cnts[1]) {
      unsigned pos = atomicAdd(&cnts[0], 1u);
      if (pos < (unsigned)CAP)
        cand[pos] = ((unsigned long long)bits << 32) |
                    (unsigned long long)(0xFFFFFFFFu - (unsigned)idx);
    }
  }
}

/* ------------------------------------------------------------------ */
__global__ __launch_bounds__(1024) void k_sort_select(
    const unsigned long long* __restrict__ cand,
    const unsigned* __restrict__ cnts,
    const float* __restrict__ boxes,
    float* __restrict__ ssc, float* __restrict__ sbox, float* __restrict__ nbox) {
  __shared__ unsigned long long keys[CAP];   /* 64 KB LDS */
  unsigned cnt = cnts[0]; if (cnt > (unsigned)CAP) cnt = CAP;
  for (int i = threadIdx.x; i < CAP; i += 1024)
    keys[i] = (i < (int)cnt) ? cand[i] : 0ull;
  __syncthreads();
  /* bitonic, descending */
  for (int k = 2; k <= CAP; k <<= 1) {
    for (int j = k >> 1; j > 0; j >>= 1) {
      for (int i = threadIdx.x; i < CAP; i += 1024) {
        int p = i ^ j;
        if (p > i) {
          unsigned long long x = keys[i], y = keys[p];
          bool desc = ((i & k) == 0);
          if (desc ? (x < y) : (x > y)) { keys[i] = y; keys[p] = x; }
        }
      }
      __syncthreads();
    }
  }
  for (int i = threadIdx.x; i < T_TOP; i += 1024) {
    unsigned long long key = keys[i];
    unsigned bits = (unsigned)(key >> 32);
    if (bits == 0u) {
      ssc[i] = 0.f;
      for (int c = 0; c < 4; ++c) { sbox[i*4+c] = 0.f; nbox[i*4+c] = 0.f; }
    } else {
      unsigned idx = 0xFFFFFFFFu - (unsigned)(key & 0xFFFFFFFFull);
      ssc[i] = __uint_as_float(bits);
      unsigned r = idx / K_CLS, c = idx % K_CLS;
      const float* bp = boxes + (size_t)r * 4;
      float x1 = fminf(fmaxf(bp[0], 0.f), IMG_W);
      float y1 = fminf(fmaxf(bp[1], 0.f), IMG_H);
      float x2 = fminf(fmaxf(bp[2], 0.f), IMG_W);
      float y2 = fminf(fmaxf(bp[3], 0.f), IMG_H);
      sbox[i*4+0] = x1; sbox[i*4+1] = y1; sbox[i*4+2] = x2; sbox[i*4+3] = y2;
      float off = (float)c * CLS_OFF;
      nbox[i*4+0] = x1+off; nbox[i*4+1] = y1+off; nbox[i*4+2] = x2+off; nbox[i*4+3] = y2+off;
    }
  }
}

/* ------------------------------------------------------------------ */
/* Pair mask: TDM-stage 2048 NMS boxes (32 KB) into LDS, then per 16x16
 * tile compute area_i + area_j with v_wmma_f32_16x16x4_f32 (rank-2 outer
 * product [area,1] x [[1],[area]]) and intersections with VALU.
 * iou > t  <=>  inter > t * (areaSum - inter + eps)   (denominator > 0),
 * avoiding the 10-op IEEE divide in the hot loop.                      */
__global__ __launch_bounds__(256) void k_mask(const float* __restrict__ nbox_g,
                                              unsigned* __restrict__ sup) {
  __shared__ float4 nbox_s[T_TOP];   /* 32 KB */
  __shared__ float  area_s[T_TOP];   /*  8 KB */

  if (threadIdx.x < 32) {            /* one wave issues the DMA */
    unsigned long long ga = (unsigned long long)(uintptr_t)nbox_g;
    unsigned lds = (unsigned)(unsigned long long)(uintptr_t)&nbox_s[0];
    u32x4 g0;
    g0.x = 1u;                                   /* count=1, user descriptor */
    g0.y = lds;                                  /* lds_addr                 */
    g0.z = (unsigned)(ga & 0xFFFFFFFFull);       /* global_addr[31:0]        */
    g0.w = (unsigned)((ga >> 32) & 0x1FFFFFFull) | (2u << 30); /* addr hi + type=2 */
    i32x8 g1;
    g1[0] = 0x00020000;          /* wg_mask=0, data_size=2 (4B)              */
    g1[1] = (int)(8192u << 16);  /* tensor_dim0 = 8192 (lo16 in [63:48])     */
    g1[2] = 0x00010000;          /* tensor_dim0 hi=0, tensor_dim1 = 1        */
    g1[3] = (int)(8192u << 16);  /* tensor_dim1 hi=0, tile_dim0 = 8192       */
    g1[4] = 1;                   /* tile_dim1 = 1, tile_dim2 = 0             */
    g1[5] = 8192;                /* tensor_dim0_stride lo32                  */
    g1[6] = 0;
    g1[7] = 0;
    i32x4 z4 = (i32x4){0, 0, 0, 0};
    i32x8 z8 = (i32x8){0, 0, 0, 0, 0, 0, 0, 0};
    __builtin_amdgcn_tensor_load_to_lds(g0, g1, z4, z4, z8, 0);
    __builtin_amdgcn_s_wait_tensorcnt(0);
  }
  __syncthreads();

  for (int i = threadIdx.x; i < T_TOP; i += 256) {
    float4 b = nbox_s[i];
    area_s[i] = fmaxf(b.z - b.x, 0.f) * fmaxf(b.w - b.y, 0.f);
  }
  __syncthreads();

  const int lane = threadIdx.x & 31;
  const int gw = (blockIdx.x * 256 + (int)threadIdx.x) >> 5;
  const int nw = ((int)gridDim.x * 256) >> 5;
  const int NTILE = (T_TOP / 16) * (T_TOP / 16);
  const bool lo = lane < 16;
  const int l16 = lane & 15;

  for (int t = gw; t < NTILE; t += nw) {
    int ti = t >> 7, tj = t & 127;
    /* A (16x4): col0 = area_i, col1 = 1, cols 2,3 = 0  -> lanes>=16 hold K=2,3 (zeros) */
    v2f a, b;
    a.x = lo ? area_s[ti * 16 + l16] : 0.f;
    a.y = lo ? 1.0f : 0.f;
    /* B (4x16): row0 = 1, row1 = area_j, rows 2,3 = 0 */
    b.x = lo ? 1.0f : 0.f;
    b.y = lo ? area_s[tj * 16 + l16] : 0.f;
    v8f c = {};
    v8f d = __builtin_amdgcn_wmma_f32_16x16x4_f32(false, a, false, b, (short)0, c, false, false);

    int jcol = tj * 16 + l16;
    float4 bj = nbox_s[jcol];
    int ibase = ti * 16 + ((lane >> 4) << 3);   /* +8 rows for lanes 16..31 */
#pragma unroll
    for (int v = 0; v < 8; ++v) {
      int i = ibase + v;
      float4 bi = nbox_s[i];
      float ix = fminf(bi.z, bj.z) - fmaxf(bi.x, bj.x);
      float iy = fminf(bi.w, bj.w) - fmaxf(bi.y, bj.y);
      float inter = fmaxf(ix, 0.f) * fmaxf(iy, 0.f);
      /* iou > NMS_T  <=>  inter > NMS_T * (sumArea - inter + eps) */
      bool p = (inter > NMS_T * (d[v] - inter + 1e-9f)) && (jcol > i);
      unsigned m = __builtin_amdgcn_ballot_w32(p);
      if (lane == 0) {
        sup[(ti * 16 + v)     * 128 + tj] = m & 0xFFFFu;  /* row v,   cols tj*16.. */
        sup[(ti * 16 + v + 8) * 128 + tj] = m >> 16;      /* row v+8              */
      }
    }
  }
}

/* ------------------------------------------------------------------ */
__global__ __launch_bounds__(128) void k_scan(const float* __restrict__ ssc,
                                              const float* __restrict__ sbox,
                                              const unsigned* __restrict__ sup,
                                              float* __restrict__ out) {
  __shared__ unsigned removed[128];        /* 128 x 16-col words = 2048 bits */
  __shared__ unsigned char keepf[T_TOP];
  int tid = threadIdx.x;
  removed[tid] = 0u;
  __syncthreads();
  for (int i = 0; i < T_TOP; ++i) {
    unsigned w = removed[i >> 4];
    __syncthreads();
    bool kp = (ssc[i] > SCORE_T) && !((w >> (i & 15)) & 1u);
    if (tid == 0) keepf[i] = kp ? 1 : 0;
    if (kp) removed[tid] |= sup[i * 128 + tid];
    __syncthreads();
  }
  for (int i = tid; i < T_TOP; i += 128) {
    bool kp = keepf[i] != 0;
    out[i * 5 + 0] = kp ? sbox[i * 4 + 0] : 0.f;
    out[i * 5 + 1] = kp ? sbox[i * 4 + 1] : 0.f;
    out[i * 5 + 2] = kp ? sbox[i * 4 + 2] : 0.f;
    out[i * 5 + 3] = kp ? sbox[i * 4 + 3] : 0.f;
    out[i * 5 + 4] = kp ? ssc[i] : 0.f;
  }
}

/* ------------------------------------------------------------------ */
extern "C" void kernel_launch(void* const* d_in, const int* in_sizes, int n_in,
                              void* d_out, int out_size, void* d_ws, size_t ws_size,
                              hipStream_t stream) {
  const float* boxes  = (const float*)d_in[0];
  const float* scores = (const float*)d_in[1];
  const float* center = (const float*)d_in[2];
  char* ws = (char*)d_ws;
  unsigned*            hist = (unsigned*)(ws + WS_HIST);
  unsigned*            cnts = (unsigned*)(ws + WS_CNT);
  unsigned long long*  cand = (unsigned long long*)(ws + WS_CAND);
  float*               ssc  = (float*)(ws + WS_SSC);
  float*               sbox = (float*)(ws + WS_SBOX);
  float*               nbox = (float*)(ws + WS_NBOX);
  unsigned*            sup  = (unsigned*)(ws + WS_SUP);
  float* out = (float*)d_out;

  k_zero<<<(HBINS + 16 + 255) / 256, 256, 0, stream>>>(hist, HBINS + 16);
  k_hist<<<(NCAND + 255) / 256, 256, 0, stream>>>(scores, center, hist);
  k_thresh<<<1, 1024, 0, stream>>>(hist, cnts);
  k_gather<<<(NCAND + 255) / 256, 256, 0, stream>>>(scores, center, cnts, cand);
  k_sort_select<<<1, 1024, 0, stream>>>(cand, cnts, boxes, ssc, sbox, nbox);
  k_mask<<<256, 256, 0, stream>>>(nbox, sup);
  k_scan<<<1, 128, 0, stream>>>(ssc, sbox, sup, out);
}